// GraphAttentionLayer_86912958202409
// MI455X (gfx1250) — compile-verified
//
#include <hip/hip_runtime.h>
#include <hip/hip_bf16.h>
#include <math.h>
#include <stdint.h>

// Problem constants (from reference): B=8, N=1024, D=256, H=8, HD=32
#define GB 8
#define GN 1024
#define GD 256
#define GH 8
#define GHD 32

typedef __attribute__((ext_vector_type(2))) float v2f;
typedef __attribute__((ext_vector_type(8))) float v8f;
typedef __attribute__((ext_vector_type(4))) unsigned int v4u;
typedef __attribute__((ext_vector_type(8))) int v8i;
typedef __attribute__((ext_vector_type(4))) int v4i;

#if __has_builtin(__builtin_amdgcn_tensor_load_to_lds)
#define HAVE_TDM 1
#endif

// -----------------------------------------------------------------------------
// Kernel 1: h = x @ W^T + b, written in [B, H, N, HD] layout.
// One block per 16-row m-tile. The 16x256 x slice (16 KB, contiguous rows) is
// staged into LDS once via the Tensor Data Mover (tensor_load_to_lds +
// s_wait_tensorcnt), then 8 waves each compute two 16x16 o-tiles with
// V_WMMA_F32_16X16X4_F32, sharing the LDS A operand (one ds_load_b64 feeds
// two WMMAs). This halves the global read traffic vs. per-wave A fetches.
// -----------------------------------------------------------------------------
__global__ __launch_bounds__(256) void gat_gemm_h(
    const float* __restrict__ x,      // [B*N, D]
    const float* __restrict__ W,      // [D, D] (h[m,o] = sum_d x[m,d]*W[o,d])
    const float* __restrict__ bias,   // [D]
    float* __restrict__ h) {          // [B, H, N, HD]
  __shared__ float Xs[16 * GD];       // 16 KB A tile

  const int lane  = threadIdx.x & 31;
  const int wave  = threadIdx.x >> 5;
  const int m0    = blockIdx.x * 16;  // 512 m-tiles
  const int l15   = lane & 15;
  const int khalf = lane >> 4;

#ifdef HAVE_TDM
  // --- Stage x rows [m0, m0+16) into Xs with the TDM (one wave issues). ---
  if (wave == 0) {
    const uint64_t ga  = (uint64_t)(uintptr_t)(x + (size_t)m0 * GD);
    const unsigned lds = (unsigned)(uintptr_t)(&Xs[0]);  // LDS byte offset
    // D# group0: count=1 | lds_addr | global_addr[56:0] | type=2
    v4u g0 = {1u, lds, (unsigned)ga,
              (unsigned)((ga >> 32) & 0x01FFFFFFu) | (2u << 30)};
    // D# group1: data_size=4B; tensor_dim0=256, tensor_dim1=16;
    // tile_dim0=256, tile_dim1=16; tensor_dim0_stride=256.
    v8i g1 = {(int)0x00020000,         // workgroup_mask=0, data_size=2 (4B)
              (int)(256u << 16),       // tensor_dim0[15:0] << 16
              (int)(16u << 16),        // tensor_dim1[15:0] << 16
              (int)(256u << 16),       // tile_dim0 << 16
              16,                      // tile_dim1=16, tile_dim2=0
              256,                     // tensor_dim0_stride[31:0]
              0, 0};
    v4i gz = {0, 0, 0, 0};
#if defined(__clang_major__) && __clang_major__ >= 23
    v8i gz8 = {0, 0, 0, 0, 0, 0, 0, 0};
    __builtin_amdgcn_tensor_load_to_lds(g0, g1, gz, gz, gz8, 0);
#else
    __builtin_amdgcn_tensor_load_to_lds(g0, g1, gz, gz, 0);
#endif
    __builtin_amdgcn_s_wait_tensorcnt(0);
  }
#else
  // Fallback: cooperative vectorized copy.
  {
    const float4* src = (const float4*)(x + (size_t)m0 * GD);
    float4* dst = (float4*)Xs;
    for (int idx = threadIdx.x; idx < 16 * GD / 4; idx += 256) dst[idx] = src[idx];
  }
#endif
  __syncthreads();

  // Wave computes o-tiles o0a = wave*16 and o0b = (wave+8)*16.
  const int o0a = wave * 16;
  const int o0b = (wave + 8) * 16;
  const float* wra = W + (size_t)(o0a + l15) * GD;
  const float* wrb = W + (size_t)(o0b + l15) * GD;
  const float* xsr = Xs + l15 * GD;

  v8f acc0 = {};
  v8f acc1 = {};
  for (int k0 = 0; k0 < GD; k0 += 4) {
    const int kb = k0 + khalf * 2;
    v2f av = *(const v2f*)(xsr + kb);      // ds_load_b64 (shared A)
    v2f b0 = *(const v2f*)(wra + kb);
    v2f b1 = *(const v2f*)(wrb + kb);
    acc0 = __builtin_amdgcn_wmma_f32_16x16x4_f32(
        false, av, false, b0, (short)0, acc0, false, false);
    acc1 = __builtin_amdgcn_wmma_f32_16x16x4_f32(
        false, av, false, b1, (short)0, acc1, false, false);
  }

  // C/D layout: VGPR vg -> row m0 + vg + 8*khalf, col o0 + (lane&15)
#pragma unroll
  for (int tsel = 0; tsel < 2; ++tsel) {
    const v8f acc = tsel ? acc1 : acc0;
    const int o   = (tsel ? o0b : o0a) + l15;
    const float bv = bias[o];
    const int hh = o >> 5;
    const int d  = o & 31;
#pragma unroll
    for (int vg = 0; vg < 8; ++vg) {
      const int gm = m0 + vg + 8 * khalf;       // global token row
      const int bb = gm >> 10;                  // batch
      const int nn = gm & (GN - 1);             // token
      h[(((size_t)(bb * GH + hh) * GN) + nn) * GHD + d] = acc[vg] + bv;
    }
  }
}

// -----------------------------------------------------------------------------
// Kernel 2: s_i[b,h,n] = h[b,h,n,:].a[b,0:32], s_j with a[b,32:64].
// Faithful to reference: `a` is indexed by the BATCH axis (B == H == 8).
// -----------------------------------------------------------------------------
__global__ __launch_bounds__(256) void gat_scores(
    const float* __restrict__ h,   // [B,H,N,HD]
    const float* __restrict__ a,   // [H, 2*HD] indexed by batch
    float* __restrict__ si, float* __restrict__ sj) {
  const int idx = blockIdx.x * 256 + threadIdx.x;   // (b*H+hh)*N + n
  const int b   = idx >> 13;                        // / (H*N)
  const float4* hp4 = (const float4*)(h + (size_t)idx * GHD);
  const float4* a0  = (const float4*)(a + b * (2 * GHD));
  const float4* a1  = (const float4*)(a + b * (2 * GHD) + GHD);
  float s0 = 0.f, s1 = 0.f;
#pragma unroll
  for (int q = 0; q < GHD / 4; ++q) {
    const float4 v = hp4[q];
    const float4 c0 = a0[q];
    const float4 c1 = a1[q];
    s0 += v.x * c0.x + v.y * c0.y + v.z * c0.z + v.w * c0.w;
    s1 += v.x * c1.x + v.y * c1.y + v.z * c1.z + v.w * c1.w;
  }
  si[idx] = s0;
  sj[idx] = s1;
}

// -----------------------------------------------------------------------------
// Kernel 3: per (b, 16-row block): for each head, masked-LeakyReLU softmax
// row-block P[16][1024] in LDS, then out += (P @ H_h)/rowsum via f32 WMMA,
// reduced across waves with ds_add_f32 into a residual-initialized F[16][256].
// adj / s_j streams are read fully lane-coalesced (wave owns 2 rows, lanes
// sweep consecutive j); raw scores live in registers; row max/sum use
// __shfl_xor wave reductions (wave32).
// Epilogue: torch-style LayerNorm (unbiased var, /(std+eps)) * gamma + beta.
// -----------------------------------------------------------------------------
__global__ __launch_bounds__(256) void gat_attn(
    const float* __restrict__ x,      // [B,N,D] residual
    const int*   __restrict__ adj,    // [B,N,N]
    const float* __restrict__ h,      // [B,H,N,HD]
    const float* __restrict__ si,     // [B,H,N]
    const float* __restrict__ sj,     // [B,H,N]
    const float* __restrict__ gamma,  // [D]
    const float* __restrict__ beta,   // [D]
    float* __restrict__ out) {        // [B,N,D]
  __shared__ float P[16 * GN];        // 64 KB prob tile (one head at a time)
  __shared__ float F[16 * GD];        // 16 KB fused output rows
  __shared__ float tmp1[256];
  __shared__ float tmp2[256];
  __shared__ float redsum[16];        // LN: row sum
  __shared__ float redsq[16];         // LN: row sumsq
  __shared__ float rowsum[16];        // softmax denominator

  const int t    = threadIdx.x;
  const int lane = t & 31;
  const int wave = t >> 5;
  const int b    = blockIdx.x >> 6;          // batch
  const int i0   = (blockIdx.x & 63) * 16;   // row block start

  const int l15   = lane & 15;
  const int khalf = lane >> 4;

  // Initialize F with the residual x rows (16 KB contiguous, float4 copy).
  {
    const float4* xr4 = (const float4*)(x + ((size_t)(b * GN) + i0) * GD);
    float4* F4 = (float4*)F;
    for (int idx = t; idx < 16 * GD / 4; idx += 256) F4[idx] = xr4[idx];
  }
  __syncthreads();

  for (int hh = 0; hh < GH; ++hh) {
    const float* sib = si + (size_t)(b * GH + hh) * GN;
    const float* sjb = sj + (size_t)(b * GH + hh) * GN;

    // ---- Softmax rows: wave owns rows 2*wave, 2*wave+1; lanes sweep j. ----
    for (int rr = 0; rr < 2; ++rr) {
      const int r = wave * 2 + rr;
      const float sir = sib[i0 + r];
      const int* adjr = adj + ((size_t)(b * GN) + i0 + r) * GN;
      float sc[32];
      float lmax = -3.0e38f;
#pragma unroll
      for (int q = 0; q < 32; ++q) {
        const int j = q * 32 + lane;       // coalesced 128B per q
        float s = sir + sjb[j];
        s = s >= 0.f ? s : 0.2f * s;       // LeakyReLU(0.2)
        if (adjr[j] == 0) s -= 1.0e9f;     // adjacency mask
        sc[q] = s;
        lmax = fmaxf(lmax, s);
      }
#pragma unroll
      for (int off = 16; off > 0; off >>= 1)
        lmax = fmaxf(lmax, __shfl_xor(lmax, off, 32));
      float lsum = 0.f;
#pragma unroll
      for (int q = 0; q < 32; ++q) {
        const float p = __expf(sc[q] - lmax);
        P[r * GN + q * 32 + lane] = p;
        lsum += p;
      }
#pragma unroll
      for (int off = 16; off > 0; off >>= 1)
        lsum += __shfl_xor(lsum, off, 32);
      if (lane == 0) rowsum[r] = lsum;
    }
    __syncthreads();

    // ---- GEMM: out[16][32] = P[16][1024] @ H_h[1024][32]; K over 8 waves. --
    const float* Hh = h + (size_t)(b * GH + hh) * GN * GHD;
    v8f acc0 = {};   // cols d = 0..15
    v8f acc1 = {};   // cols d = 16..31
    const int ks0 = wave * 128;
    for (int ks = 0; ks < 128; ks += 4) {
      const int kb = ks0 + ks + khalf * 2;
      // A tile from LDS (8B-aligned pair along K)
      v2f av = *(const v2f*)(&P[l15 * GN + kb]);
      // B tile: B[k][n] = Hh[k][d]; lane holds K = kb, kb+1
      const float* hb = Hh + (size_t)kb * GHD + l15;
      v2f b0 = {hb[0],  hb[GHD]};         // d = l15
      v2f b1 = {hb[16], hb[GHD + 16]};    // d = 16 + l15
      acc0 = __builtin_amdgcn_wmma_f32_16x16x4_f32(
          false, av, false, b0, (short)0, acc0, false, false);
      acc1 = __builtin_amdgcn_wmma_f32_16x16x4_f32(
          false, av, false, b1, (short)0, acc1, false, false);
    }

    // Scale by 1/rowsum (softmax denominator), reduce across waves into F.
    const int col0 = hh * GHD + l15;
#pragma unroll
    for (int vg = 0; vg < 8; ++vg) {
      const int rr = vg + 8 * khalf;
      const float invl = 1.0f / rowsum[rr];
      atomicAdd(&F[rr * GD + col0],      acc0[vg] * invl);  // ds_add_f32
      atomicAdd(&F[rr * GD + col0 + 16], acc1[vg] * invl);
    }
    __syncthreads();  // also protects P reuse by next head
  }

  // ---- Epilogue: LayerNorm (torch unbiased std), gamma/beta, store. -------
  {
    const int r  = t >> 4;
    const int c0 = (t & 15) * 16;
    float s = 0.f, s2 = 0.f;
#pragma unroll 4
    for (int c = 0; c < 16; ++c) {
      const float v = F[r * GD + c0 + c];
      s += v;
      s2 += v * v;
    }
    tmp1[t] = s;
    tmp2[t] = s2;
    __syncthreads();
    if (t < 16) {
      float a0 = 0.f, a1 = 0.f;
      for (int q = 0; q < 16; ++q) { a0 += tmp1[t * 16 + q]; a1 += tmp2[t * 16 + q]; }
      redsum[t] = a0;
      redsq[t]  = a1;
    }
    __syncthreads();
    const float mean = redsum[r] * (1.0f / GD);
    const float varb = redsq[r] * (1.0f / GD) - mean * mean;
    const float var  = varb * ((float)GD / (float)(GD - 1));   // unbiased
    const float inv  = 1.0f / (sqrtf(var) + 1e-6f);
    float* orow = out + ((size_t)(b * GN) + i0 + r) * GD;
#pragma unroll 4
    for (int c = c0; c < c0 + 16; ++c) {
      const float v = (F[r * GD + c] - mean) * inv;
      orow[c] = v * gamma[c] + beta[c];
    }
  }
}

// -----------------------------------------------------------------------------
// Launch
// Inputs (setup_inputs order): x, adj, W_w, W_b, a, gamma, beta
// Workspace: h [B,H,N,HD] (8 MB) + s_i (256 KB) + s_j (256 KB)
// -----------------------------------------------------------------------------
extern "C" void kernel_launch(void* const* d_in, const int* in_sizes, int n_in,
                              void* d_out, int out_size, void* d_ws, size_t ws_size,
                              hipStream_t stream) {
  const float* x     = (const float*)d_in[0];
  const int*   adj   = (const int*)  d_in[1];
  const float* W_w   = (const float*)d_in[2];
  const float* W_b   = (const float*)d_in[3];
  const float* a     = (const float*)d_in[4];
  const float* gamma = (const float*)d_in[5];
  const float* beta  = (const float*)d_in[6];
  float* out = (float*)d_out;

  float* h  = (float*)d_ws;                         // B*H*N*HD = 2,097,152 f32
  float* si = h + (size_t)GB * GH * GN * GHD;       // 65,536 f32
  float* sj = si + (size_t)GB * GH * GN;            // 65,536 f32

  // One block per 16-row m-tile; TDM-staged A tile; 16 o-tiles over 8 waves.
  gat_gemm_h<<<512, 256, 0, stream>>>(x, W_w, W_b, h);
  // 65536 (b,h,n) score dots
  gat_scores<<<256, 256, 0, stream>>>(h, a, si, sj);
  // one block per (b, 16-row block)
  gat_attn<<<GB * (GN / 16), 256, 0, stream>>>(x, adj, h, si, sj, gamma, beta, out);
}